// SSMCore_79577154060422
// MI455X (gfx1250) — compile-verified
//
#include <hip/hip_runtime.h>
#include <hip/hip_bf16.h>
#include <math.h>

// ---- Problem constants (from reference) ----
#define BATCH   2
#define TLEN    1024
#define DIN     1024
#define DSTATE  16
#define KCONV   4
#define MROWS   (BATCH * TLEN)   // 2048 GEMM rows
#define NC      8                // scan chunks along T
#define CT      (TLEN / NC)      // 128 timesteps per chunk

typedef __attribute__((ext_vector_type(2))) float v2f;
typedef __attribute__((ext_vector_type(8))) float v8f;

__device__ __forceinline__ float softplus_f(float x) {
    return (x > 20.0f) ? x : log1pf(__expf(x));
}

// ---- CDNA5 async global<->LDS DMA helpers (ASYNCcnt-tracked, GV mode) ----
__device__ __forceinline__ uint32_t lds_off_of(const void* p) {
    // LDS aperture: low 32 bits of a flat pointer are the LDS byte address
    return (uint32_t)(uintptr_t)p;
}
__device__ __forceinline__ void async_load_b128(uint32_t lds, uint64_t gaddr) {
    asm volatile("global_load_async_to_lds_b128 %0, %1, off"
                 :: "v"(lds), "v"(gaddr) : "memory");
}
__device__ __forceinline__ void async_store_b128(uint64_t gaddr, uint32_t lds) {
    asm volatile("global_store_async_from_lds_b128 %0, %1, off"
                 :: "v"(gaddr), "v"(lds) : "memory");
}
__device__ __forceinline__ void wait_async0() {
    asm volatile("s_wait_asynccnt 0" ::: "memory");
}

// ---------------------------------------------------------------------------
// Kernel 1: causal depthwise conv1d (K=4, left-pad 3) + bias + SiLU -> xc
// ---------------------------------------------------------------------------
__global__ void conv_silu_kernel(const float* __restrict__ x,
                                 const float* __restrict__ w,
                                 const float* __restrict__ bias,
                                 float* __restrict__ xc) {
    int idx = blockIdx.x * blockDim.x + threadIdx.x;
    int d  = idx & (DIN - 1);
    int bt = idx >> 10;
    int t  = bt & (TLEN - 1);
    float acc = bias[d];
#pragma unroll
    for (int k = 0; k < KCONV; ++k) {
        int tt = t - (KCONV - 1) + k;
        if (tt >= 0)
            acc = fmaf(x[(bt - t + tt) * DIN + d], w[d * KCONV + k], acc);
    }
    float sg = 1.0f / (1.0f + __expf(-acc));
    xc[idx] = acc * sg;
}

// ---------------------------------------------------------------------------
// Kernel 2: proj = xc(2048x1024) @ x_proj_w^T via V_WMMA_F32_16X16X4_F32
// ---------------------------------------------------------------------------
__global__ void proj_wmma_kernel(const float* __restrict__ xc,
                                 const float* __restrict__ W,
                                 float* __restrict__ proj) {
    int lane = threadIdx.x & 31;
    int tile = blockIdx.x * (blockDim.x >> 5) + (threadIdx.x >> 5);
    int mT = tile >> 1;
    int nT = tile & 1;
    int rowA = mT * 16 + (lane & 15);
    int nB   = nT * 16 + (lane & 15);
    int kh   = (lane >> 4) * 2;

    v8f acc = {};
    for (int k0 = 0; k0 < DIN; k0 += 4) {
        v2f a, b;
        a.x = xc[rowA * DIN + k0 + kh];
        a.y = xc[rowA * DIN + k0 + kh + 1];
        b.x = W[nB * DIN + k0 + kh];
        b.y = W[nB * DIN + k0 + kh + 1];
        acc = __builtin_amdgcn_wmma_f32_16x16x4_f32(
            false, a, false, b, (short)0, acc, false, false);
    }
    int cm = mT * 16 + (lane >> 4) * 8;
#pragma unroll
    for (int i = 0; i < 8; ++i)
        proj[(cm + i) * (2 * DSTATE) + nB] = acc[i];
}

// ---------------------------------------------------------------------------
// Kernel 3: dt = softplus(softplus(B_ssm @ dt_proj_w^T + b)), WMMA f32 K=16
// ---------------------------------------------------------------------------
__global__ void dt_wmma_kernel(const float* __restrict__ proj,
                               const float* __restrict__ dtw,
                               const float* __restrict__ dtb,
                               float* __restrict__ dt) {
    int lane = threadIdx.x & 31;
    int tile = blockIdx.x * (blockDim.x >> 5) + (threadIdx.x >> 5);
    int mT = tile >> 6;
    int nT = tile & 63;
    int rowA = mT * 16 + (lane & 15);
    int nB   = nT * 16 + (lane & 15);
    int kh   = (lane >> 4) * 2;

    v8f acc = {};
#pragma unroll
    for (int k0 = 0; k0 < DSTATE; k0 += 4) {
        v2f a, b;
        a.x = proj[rowA * (2 * DSTATE) + k0 + kh];
        a.y = proj[rowA * (2 * DSTATE) + k0 + kh + 1];
        b.x = dtw[nB * DSTATE + k0 + kh];
        b.y = dtw[nB * DSTATE + k0 + kh + 1];
        acc = __builtin_amdgcn_wmma_f32_16x16x4_f32(
            false, a, false, b, (short)0, acc, false, false);
    }
    int cm = mT * 16 + (lane >> 4) * 8;
    float bv = dtb[nB];
#pragma unroll
    for (int i = 0; i < 8; ++i) {
        float v = acc[i] + bv;
        v = softplus_f(softplus_f(v));
        dt[(cm + i) * DIN + nB] = v;
    }
}

// ---------------------------------------------------------------------------
// Scan pass 1: per (b, 16-channel group, chunk) compute local scan end-state
// Hend and chunk decay product P, starting from h=0. Inputs staged to LDS via
// CDNA5 async global->LDS DMA; inner loop is pure LDS + VALU.
// ---------------------------------------------------------------------------
__global__ void scan_pass1_kernel(const float* __restrict__ xc,
                                  const float* __restrict__ proj,
                                  const float* __restrict__ dt,
                                  float* __restrict__ Hend,
                                  float* __restrict__ Pprod) {
    __shared__ __align__(16) float s_dt[CT * 16];
    __shared__ __align__(16) float s_xc[CT * 16];
    __shared__ __align__(16) float s_pj[CT * 32];

    int tid = threadIdx.x;
    int blk = blockIdx.x;                 // 0..1023
    int c = blk & (NC - 1);
    int g = (blk >> 3) & 63;
    int b = blk >> 9;
    int d0 = g * 16;
    int rbase = b * TLEN + c * CT;

    uint64_t g_dt = (uint64_t)(uintptr_t)(dt   + (size_t)rbase * DIN + d0);
    uint64_t g_xc = (uint64_t)(uintptr_t)(xc   + (size_t)rbase * DIN + d0);
    uint64_t g_pj = (uint64_t)(uintptr_t)(proj + (size_t)rbase * (2 * DSTATE));

    // dt/xc tiles: CT rows x 16 floats (64B rows) = CT*4 b128 transfers each
    for (int l = tid; l < CT * 4; l += 256) {
        int trow = l >> 2, j = l & 3;
        uint64_t off = (uint64_t)trow * (DIN * 4) + (uint64_t)j * 16;
        async_load_b128(lds_off_of(&s_dt[trow * 16 + j * 4]), g_dt + off);
        async_load_b128(lds_off_of(&s_xc[trow * 16 + j * 4]), g_xc + off);
    }
    // proj tile: CT rows x 32 floats, globally contiguous (16KB)
    for (int l = tid; l < CT * 8; l += 256) {
        async_load_b128(lds_off_of(&s_pj[l * 4]), g_pj + (uint64_t)l * 16);
    }
    wait_async0();
    __syncthreads();

    int s = tid & 15, ch = tid >> 4;
    const float Aneg = -(float)(s + 1);
    const float EPS = 1e-38f;
    float h = 0.0f, P = 1.0f;
    for (int tt = 0; tt < CT; ++tt) {
        float dtv = s_dt[tt * 16 + ch];        // broadcast within 16-lane group
        float xcv = s_xc[tt * 16 + ch];
        float bs  = s_pj[tt * 32 + s];
        float dA  = fmaxf(__expf(dtv * Aneg), EPS);
        float dBx = fmaxf(dtv * bs * xcv, EPS);
        h = fmaf(dA, h, dBx);
        P *= dA;
    }
    int d = d0 + ch;
    size_t q = ((size_t)(b * DIN + d) * DSTATE + s) * NC + c;
    Hend[q]  = h;
    Pprod[q] = P;
}

// ---------------------------------------------------------------------------
// Scan combine: per (b,d,s) prefix-compose the 8 chunk summaries.
//   Hinit[c] = H_{c-1};  H_c = H_{c-1}*P_c + Hend_c
// ---------------------------------------------------------------------------
__global__ void scan_combine_kernel(const float* __restrict__ Hend,
                                    const float* __restrict__ Pprod,
                                    float* __restrict__ Hinit) {
    size_t q = (size_t)blockIdx.x * blockDim.x + threadIdx.x;  // 0..32767
    const float* He = Hend  + q * NC;
    const float* Pp = Pprod + q * NC;
    float* Hi = Hinit + q * NC;
    float acc = 0.0f;
#pragma unroll
    for (int c = 0; c < NC; ++c) {
        Hi[c] = acc;
        acc = fmaf(acc, Pp[c], He[c]);
    }
}

// ---------------------------------------------------------------------------
// Scan pass 2: re-scan each chunk seeded with Hinit, contract with C over the
// 16-lane state group (shfl_xor tree), stage y tile in LDS, drain with
// async LDS->global stores (fully coalesced 64B rows).
// ---------------------------------------------------------------------------
__global__ void scan_pass2_kernel(const float* __restrict__ xc,
                                  const float* __restrict__ proj,
                                  const float* __restrict__ dt,
                                  const float* __restrict__ Hinit,
                                  const float* __restrict__ Dvec,
                                  float* __restrict__ y) {
    __shared__ __align__(16) float s_dt[CT * 16];
    __shared__ __align__(16) float s_xc[CT * 16];
    __shared__ __align__(16) float s_pj[CT * 32];
    __shared__ __align__(16) float s_y [CT * 16];

    int tid = threadIdx.x;
    int blk = blockIdx.x;
    int c = blk & (NC - 1);
    int g = (blk >> 3) & 63;
    int b = blk >> 9;
    int d0 = g * 16;
    int rbase = b * TLEN + c * CT;

    uint64_t g_dt = (uint64_t)(uintptr_t)(dt   + (size_t)rbase * DIN + d0);
    uint64_t g_xc = (uint64_t)(uintptr_t)(xc   + (size_t)rbase * DIN + d0);
    uint64_t g_pj = (uint64_t)(uintptr_t)(proj + (size_t)rbase * (2 * DSTATE));

    for (int l = tid; l < CT * 4; l += 256) {
        int trow = l >> 2, j = l & 3;
        uint64_t off = (uint64_t)trow * (DIN * 4) + (uint64_t)j * 16;
        async_load_b128(lds_off_of(&s_dt[trow * 16 + j * 4]), g_dt + off);
        async_load_b128(lds_off_of(&s_xc[trow * 16 + j * 4]), g_xc + off);
    }
    for (int l = tid; l < CT * 8; l += 256) {
        async_load_b128(lds_off_of(&s_pj[l * 4]), g_pj + (uint64_t)l * 16);
    }
    wait_async0();
    __syncthreads();

    int s = tid & 15, ch = tid >> 4;
    int d = d0 + ch;
    const float Aneg = -(float)(s + 1);
    const float EPS = 1e-38f;
    const float Dd = Dvec[d];
    float h = Hinit[((size_t)(b * DIN + d) * DSTATE + s) * NC + c];

    for (int tt = 0; tt < CT; ++tt) {
        float dtv = s_dt[tt * 16 + ch];
        float xcv = s_xc[tt * 16 + ch];
        float bs  = s_pj[tt * 32 + s];
        float cs  = s_pj[tt * 32 + 16 + s];
        float dA  = fmaxf(__expf(dtv * Aneg), EPS);
        float dBx = fmaxf(dtv * bs * xcv, EPS);
        h = fmaf(dA, h, dBx);
        float v = cs * h;
        v += __shfl_xor(v, 1, 16);
        v += __shfl_xor(v, 2, 16);
        v += __shfl_xor(v, 4, 16);
        v += __shfl_xor(v, 8, 16);
        if (s == 0)
            s_y[tt * 16 + ch] = fmaf(Dd, xcv, v);
    }
    __syncthreads();

    uint64_t g_y = (uint64_t)(uintptr_t)(y + (size_t)rbase * DIN + d0);
    for (int l = tid; l < CT * 4; l += 256) {
        int trow = l >> 2, j = l & 3;
        uint64_t off = (uint64_t)trow * (DIN * 4) + (uint64_t)j * 16;
        async_store_b128(g_y + off, lds_off_of(&s_y[trow * 16 + j * 4]));
    }
    wait_async0();
}

// ---------------------------------------------------------------------------
extern "C" void kernel_launch(void* const* d_in, const int* in_sizes, int n_in,
                              void* d_out, int out_size, void* d_ws, size_t ws_size,
                              hipStream_t stream) {
    (void)in_sizes; (void)n_in; (void)out_size; (void)ws_size;

    const float* x        = (const float*)d_in[0];
    const float* conv_w   = (const float*)d_in[1];
    const float* conv_b   = (const float*)d_in[2];
    const float* x_proj_w = (const float*)d_in[3];
    const float* dt_proj_w= (const float*)d_in[4];
    const float* dt_proj_b= (const float*)d_in[5];
    const float* Dvec     = (const float*)d_in[6];
    float* out = (float*)d_out;

    float* ws    = (float*)d_ws;
    float* xc    = ws;                                     // 2M floats (8 MB)
    float* proj  = xc    + (size_t)BATCH * TLEN * DIN;     // 64K floats (256 KB)
    float* dt    = proj  + (size_t)MROWS * 2 * DSTATE;     // 2M floats (8 MB)
    float* Hend  = dt    + (size_t)BATCH * TLEN * DIN;     // 256K floats (1 MB)
    float* Pprod = Hend  + (size_t)BATCH * DIN * DSTATE * NC;
    float* Hinit = Pprod + (size_t)BATCH * DIN * DSTATE * NC;

    // 1) conv + SiLU
    conv_silu_kernel<<<(BATCH * TLEN * DIN) / 256, 256, 0, stream>>>(
        x, conv_w, conv_b, xc);
    // 2) proj GEMM (WMMA f32): 256 wave-tiles
    proj_wmma_kernel<<<32, 256, 0, stream>>>(xc, x_proj_w, proj);
    // 3) dt GEMM (WMMA f32) + bias + double softplus: 8192 wave-tiles
    dt_wmma_kernel<<<1024, 256, 0, stream>>>(proj, dt_proj_w, dt_proj_b, dt);
    // 4) chunked scan: pass1 (local scans) -> combine -> pass2 (seeded + output)
    scan_pass1_kernel<<<BATCH * (DIN / 16) * NC, 256, 0, stream>>>(
        xc, proj, dt, Hend, Pprod);
    scan_combine_kernel<<<(BATCH * DIN * DSTATE) / 256, 256, 0, stream>>>(
        Hend, Pprod, Hinit);
    scan_pass2_kernel<<<BATCH * (DIN / 16) * NC, 256, 0, stream>>>(
        xc, proj, dt, Hinit, Dvec, out);
}